// hpCNNEmbedding_81638738362866
// MI455X (gfx1250) — compile-verified
//
#include <hip/hip_runtime.h>

// CDNA5 (gfx1250) fp32 WMMA: D(16x16,f32) = A(16x4,f32) * B(4x16,f32) + C
typedef float v2f __attribute__((ext_vector_type(2)));
typedef float v8f __attribute__((ext_vector_type(8)));

__device__ __forceinline__ int imin(int a, int b) { return a < b ? a : b; }

// ---------------------------------------------------------------------------
// Naive transpose: in [R, C] -> out [C, R]. Read coalesced, write strided.
// Used once per call for the network input and all weight matrices.
// ---------------------------------------------------------------------------
__global__ void transpose_naive(const float* __restrict__ in,
                                float* __restrict__ out, int R, int C)
{
    int t = blockIdx.x * blockDim.x + threadIdx.x;
    if (t < R * C) {
        int r = t / C, c = t - r * C;
        out[c * R + r] = in[t];
    }
}

// ---------------------------------------------------------------------------
// Spherical conv block (batch-innermost layout everywhere):
//   xt  : [CIN, NPIX, 64]   element (c,n,b) at (c*NPIX+n)*64 + b
//   nbr : [9*NPIX] int32, values in [0, NPIX] (NPIX == zero pad)
//   Wt  : [CIN*9, COUT]     element (j,o) at j*COUT + o  (transposed weights)
//   out : [COUT, NPIX/4, 64]
// grid = (NPIX/4, 4 batch-tiles, ceil(COUT/16)), block = 32 (one wave).
// One wave: 16(cout) x 16(batch) tile for 4 consecutive pixels concurrently
// (4 independent accumulators sharing each A fragment), then
// bias + ReLU + average -> one pooled output tile. All guards branch-free.
// ---------------------------------------------------------------------------
template <int CIN, int COUT, int NPIX>
__global__ __launch_bounds__(32)
void sconv_block_wmma(const float* __restrict__ xt,
                      const int*   __restrict__ nbr,
                      const float* __restrict__ Wt,
                      const float* __restrict__ bias,
                      float*       __restrict__ out)
{
    constexpr int KTOT    = CIN * 9;
    constexpr int KPAD    = (KTOT + 3) & ~3;
    constexpr int NP      = NPIX / 4;
    constexpr bool KEXACT = (KTOT % 4) == 0;
    constexpr bool MFULL  = (COUT % 16) == 0;

    const int lane = threadIdx.x;
    const int half = lane >> 4;                // 0 | 1
    const int row  = lane & 15;

    const int p   = blockIdx.x;                // pooled pixel
    const int bb0 = blockIdx.y << 4;           // batch tile base
    const int m0  = blockIdx.z << 4;           // cout tile base

    const int bl  = bb0 + row;                 // batch (B frag col / store)
    const int oA  = m0 + row;                  // out-channel (A frag row)
    const int oAc = MFULL ? oA : imin(oA, COUT - 1);
    const bool oval = MFULL || (oA < COUT);

    const int nbase = 36 * p;                  // 9 * (4p)

    v8f acc[4];
    #pragma unroll
    for (int q = 0; q < 4; ++q)
        acc[q] = (v8f){0.f,0.f,0.f,0.f,0.f,0.f,0.f,0.f};

    #pragma unroll 9
    for (int kk = 0; kk < KPAD; kk += 4) {
        const int j0 = kk + (half << 1);       // lanes<16: kk,kk+1 ; >=16: kk+2,kk+3
        const int j1 = j0 + 1;
        const bool v0 = KEXACT || (j0 < KTOT);
        const bool v1 = KEXACT || (j1 < KTOT);
        const int j0c = KEXACT ? j0 : imin(j0, KTOT - 1);
        const int j1c = KEXACT ? j1 : imin(j1, KTOT - 1);

        // --- A fragment: coalesced (lane-varying o is contiguous) ----------
        v2f a;
        {
            float w0 = Wt[j0c * COUT + oAc];
            float w1 = Wt[j1c * COUT + oAc];
            a.x = (oval && v0) ? w0 : 0.0f;
            a.y = (oval && v1) ? w1 : 0.0f;
        }

        // K index -> (channel, tap)
        const int c0 = j0c / 9, k0 = j0c - 9 * c0;
        const int c1 = j1c / 9, k1 = j1c - 9 * c1;

        // --- B fragments: gather rows, batch-contiguous (coalesced) --------
        #pragma unroll
        for (int q = 0; q < 4; ++q) {
            const int idx0 = nbr[nbase + 9 * q + k0];
            const int idx1 = nbr[nbase + 9 * q + k1];
            const float xv0 = xt[(c0 * NPIX + imin(idx0, NPIX - 1)) * 64 + bl];
            const float xv1 = xt[(c1 * NPIX + imin(idx1, NPIX - 1)) * 64 + bl];
            v2f bf;
            bf.x = (v0 && idx0 < NPIX) ? xv0 : 0.0f;
            bf.y = (v1 && idx1 < NPIX) ? xv1 : 0.0f;

            acc[q] = __builtin_amdgcn_wmma_f32_16x16x4_f32(
                         false, a, false, bf, (short)0, acc[q], false, false);
        }
    }

    // --- bias + ReLU + AvgPool4 + coalesced store ---------------------------
    #pragma unroll
    for (int r = 0; r < 8; ++r) {
        const int  o  = m0 + r + (half << 3);  // C/D: M = r (+8 upper half)
        const float bv = bias[MFULL ? o : imin(o, COUT - 1)];
        float s = 0.0f;
        #pragma unroll
        for (int q = 0; q < 4; ++q)
            s += fmaxf(acc[q][r] + bv, 0.0f);
        if (MFULL || (o < COUT))
            out[(o * NP + p) * 64 + bl] = s * 0.25f;
    }
}

// ---------------------------------------------------------------------------
// FC layer as WMMA GEMM, batch-innermost:
//   Xt : [IN, 64], Wt : [IN, OUT], bias : [OUT]
//   out: TSTORE ? [OUT, 64] : [64, OUT]
// grid = (4 batch-tiles, ceil(OUT/16)), block = 32
// ---------------------------------------------------------------------------
template <int IN, int OUT, bool RELU, bool TSTORE>
__global__ __launch_bounds__(32)
void fc_wmma(const float* __restrict__ Xt,
             const float* __restrict__ Wt,
             const float* __restrict__ bias,
             float*       __restrict__ out)
{
    constexpr bool MFULL = (OUT % 16) == 0;

    const int lane = threadIdx.x;
    const int half = lane >> 4;
    const int row  = lane & 15;

    const int bb0 = blockIdx.x << 4;
    const int m0  = blockIdx.y << 4;

    const int bl  = bb0 + row;                 // < 64 always (4 exact tiles)
    const int oA  = m0 + row;
    const int oAc = MFULL ? oA : imin(oA, OUT - 1);
    const bool oval = MFULL || (oA < OUT);

    v8f acc = {0.f,0.f,0.f,0.f,0.f,0.f,0.f,0.f};

    #pragma unroll 8
    for (int kk = 0; kk < IN; kk += 4) {
        const int j0 = kk + (half << 1);
        const int j1 = j0 + 1;

        v2f a;
        {
            float w0 = Wt[j0 * OUT + oAc];     // coalesced over lanes
            float w1 = Wt[j1 * OUT + oAc];
            a.x = oval ? w0 : 0.0f;
            a.y = oval ? w1 : 0.0f;
        }
        v2f bf;
        bf.x = Xt[j0 * 64 + bl];               // coalesced over lanes
        bf.y = Xt[j1 * 64 + bl];

        acc = __builtin_amdgcn_wmma_f32_16x16x4_f32(
                  false, a, false, bf, (short)0, acc, false, false);
    }

    #pragma unroll
    for (int r = 0; r < 8; ++r) {
        const int o = m0 + r + (half << 3);
        if (MFULL || (o < OUT)) {
            float v = acc[r] + bias[o];
            v = RELU ? fmaxf(v, 0.0f) : v;
            if (TSTORE) out[o * 64 + bl] = v;      // [OUT, 64]
            else        out[bl * OUT + o] = v;     // [64, OUT] (final)
        }
    }
}

// ---------------------------------------------------------------------------
// Host-side orchestration
// Input order (setup_inputs dict order):
//   0:x, 1..6:nbr0..nbr5, 7..18:(W0,b0 .. W5,b5), 19..24:(fcW0,fcb0 .. fcW2,fcb2)
// ---------------------------------------------------------------------------
extern "C" void kernel_launch(void* const* d_in, const int* in_sizes, int n_in,
                              void* d_out, int out_size, void* d_ws, size_t ws_size,
                              hipStream_t stream)
{
    (void)in_sizes; (void)n_in; (void)out_size; (void)ws_size;

    const float* x = (const float*)d_in[0];
    const int*   nbr[6];
    const float* Wc[6];
    const float* bc[6];
    for (int i = 0; i < 6; ++i) {
        nbr[i] = (const int*)  d_in[1 + i];
        Wc[i]  = (const float*)d_in[7 + 2 * i];
        bc[i]  = (const float*)d_in[8 + 2 * i];
    }
    const float* fcW[3];
    const float* fcb[3];
    for (int j = 0; j < 3; ++j) {
        fcW[j] = (const float*)d_in[19 + 2 * j];
        fcb[j] = (const float*)d_in[20 + 2 * j];
    }

    // Workspace layout:
    //   r0  @ 0      (26 MB region)  : even-stage activations
    //   r1  @ 26 MB  (14 MB region)  : x-transpose, then odd-stage activations
    //   wts @ 40 MB  (~2.2 MB)       : transposed weights
    float* r0 = (float*)d_ws;
    float* r1 = (float*)((char*)d_ws + (26u << 20));
    float* wt = (float*)((char*)d_ws + (40u << 20));

    static const int CH[7] = {1, 8, 16, 32, 64, 128, 256};
    // Transposed conv weights Wt[i] : [CIN*9, COUT]
    float* Wt[6];
    {
        float* pos = wt;
        for (int i = 0; i < 6; ++i) {
            Wt[i] = pos;
            pos += CH[i] * 9 * CH[i + 1];
        }
        float* fWt0 = pos; pos += 3072 * 48;
        float* fWt1 = pos; pos += 48 * 48;
        float* fWt2 = pos; pos += 48 * 20;

        // Weight transposes (tiny, one pass each)
        for (int i = 0; i < 6; ++i) {
            int R = CH[i + 1], C = CH[i] * 9;
            transpose_naive<<<(R * C + 255) / 256, 256, 0, stream>>>(Wc[i], Wt[i], R, C);
        }
        transpose_naive<<<(48 * 3072 + 255) / 256, 256, 0, stream>>>(fcW[0], fWt0, 48, 3072);
        transpose_naive<<<(48 * 48 + 255) / 256, 256, 0, stream>>>(fcW[1], fWt1, 48, 48);
        transpose_naive<<<(20 * 48 + 255) / 256, 256, 0, stream>>>(fcW[2], fWt2, 20, 48);

        // Input transpose: x [64, 49152] -> xt [49152, 64] (into r1 region)
        transpose_naive<<<(64 * 49152 + 255) / 256, 256, 0, stream>>>(x, r1, 64, 49152);

        // ---- conv stack (ping-pong r1/r0; stage-1 output overwrites dead xt)
#define LAUNCH_SCONV(CIN, COUT, NPIXV, IN_PTR, OUT_PTR, LI)                    \
        sconv_block_wmma<CIN, COUT, NPIXV>                                     \
            <<<dim3(NPIXV / 4, 4, (COUT + 15) / 16), 32, 0, stream>>>(         \
                IN_PTR, nbr[LI], Wt[LI], bc[LI], OUT_PTR)

        LAUNCH_SCONV(  1,   8, 49152, r1, r0, 0);
        LAUNCH_SCONV(  8,  16, 12288, r0, r1, 1);
        LAUNCH_SCONV( 16,  32,  3072, r1, r0, 2);
        LAUNCH_SCONV( 32,  64,   768, r0, r1, 3);
        LAUNCH_SCONV( 64, 128,   192, r1, r0, 4);
        LAUNCH_SCONV(128, 256,    48, r0, r1, 5);
#undef LAUNCH_SCONV
        // r1 now holds [256, 12, 64] == Xt [3072, 64] with feature f = c*12+n
        // (matches reference h.reshape(B,-1) feature order).

        fc_wmma<3072, 48, true,  true ><<<dim3(4, 3), 32, 0, stream>>>(r1, fWt0, fcb[0], r0);
        fc_wmma<  48, 48, true,  true ><<<dim3(4, 3), 32, 0, stream>>>(r0, fWt1, fcb[1], r1);
        fc_wmma<  48, 20, false, false><<<dim3(4, 2), 32, 0, stream>>>(r1, fWt2, fcb[2],
                                                                       (float*)d_out);
    }
}